// SuperpixelClustering_2d_point_38268158608136
// MI455X (gfx1250) — compile-verified
//
#include <hip/hip_runtime.h>
#include <hip/hip_bf16.h>
#include <math.h>

// ---------------------------------------------------------------------------
// Superpixel clustering (k-means-like) for MI455X / gfx1250, wave32.
// Distance matrix D^2 = ||x||^2 - 2 x.b + ||b||^2  (b = c - eps) computed with
// V_WMMA_F32_16X16X4_F32 (two chained WMMAs cover K=8).
// Scan runs in the SQUARED-distance domain (sqrt is monotonic; only
// comparisons affect labels).  The `enough` gate (cluster has >1 in-radius
// point) is folded into the published spatial coords: gated-out centers get
// coords of 1e30 so dx^2+dy^2 -> +inf and the radius test fails, which is
// decision-identical to `in_r && enough` but branch-free in the hot scan.
// ---------------------------------------------------------------------------

#define NF   8
#define NC   128
#define EPSV 1e-6f
#define THRESH 1e-4f
#define MAX_ITERS 5
#define FARV 1.0e30f

typedef __attribute__((ext_vector_type(2))) float v2f;
typedef __attribute__((ext_vector_type(8))) float v8f;

// ----------------------------- init ----------------------------------------
__global__ void spc_init_kernel(const float* __restrict__ centers_in,
                                float* __restrict__ cen, int* __restrict__ conv) {
    int i = threadIdx.x;
    if (i < NC * NF) cen[i] = centers_in[i];
    if (i == 0) *conv = 0;
}

__global__ void spc_zero_counts_kernel(int* __restrict__ counts) {
    counts[threadIdx.x] = 0;
}

// --------------------- per-center in-radius counts --------------------------
__global__ void spc_count_kernel(const float* __restrict__ data,
                                 const float* __restrict__ cen,
                                 int* __restrict__ counts, float r2) {
    __shared__ float sC2[NC * 2];
    __shared__ int   sCnt[NC];
    int tid = threadIdx.x;
    for (int i = tid; i < NC; i += 256) {
        sC2[2 * i]     = cen[i * NF + 0];
        sC2[2 * i + 1] = cen[i * NF + 1];
        sCnt[i] = 0;
    }
    __syncthreads();
    int p = blockIdx.x * 256 + tid;
    float x0 = data[(size_t)p * NF + 0];
    float x1 = data[(size_t)p * NF + 1];
    #pragma unroll 4
    for (int i = 0; i < NC; ++i) {
        float dx = x0 - sC2[2 * i];
        float dy = x1 - sC2[2 * i + 1];
        if (dx * dx + dy * dy <= r2) atomicAdd(&sCnt[i], 1);
    }
    __syncthreads();
    for (int i = tid; i < NC; i += 256) atomicAdd(&counts[i], sCnt[i]);
}

// --- gated spatial coords, b = c-eps transposed, ||b||^2, zero accumulators --
__global__ void spc_prep_kernel(const float* __restrict__ cen,
                                const int* __restrict__ counts,
                                float* __restrict__ bT, float* __restrict__ bb,
                                float* __restrict__ g2,
                                float* __restrict__ sums, float* __restrict__ cnts) {
    int i = threadIdx.x;               // 0..127, one per center
    bool enough = counts[i] > 1;
    // gated spatial coords: far away when `enough` fails -> radius test fails
    g2[2 * i]     = enough ? cen[i * NF + 0] : FARV;
    g2[2 * i + 1] = enough ? cen[i * NF + 1] : FARV;
    float acc = 0.f;
    #pragma unroll
    for (int f = 0; f < NF; ++f) {
        float b = cen[i * NF + f] - EPSV;  // x - c + eps == x - (c - eps)
        bT[f * NC + i] = b;
        acc += b * b;
        sums[i * NF + f] = 0.f;
    }
    bb[i] = acc;
    cnts[i] = 0.f;
}

// ------------------------- WMMA assignment ----------------------------------
// Block = 256 threads = 8 wave32 waves; each wave owns a 16-point tile and
// sweeps all 128 centers as 8 WMMA n-tiles (2 chained K=4 WMMAs each).
// After each tile, the 16x16 D^2 block bounces through LDS and lanes 0-15 run
// the faithful sequential per-center scan with carried (dval^2, label).
__global__ void spc_assign_kernel(const float* __restrict__ data,
                                  const float* __restrict__ g2,
                                  const float* __restrict__ bT,
                                  const float* __restrict__ bb,
                                  int* __restrict__ tmp_labels, float r2) {
    __shared__ float sBT[NF * NC];     // b transposed: [f][n]
    __shared__ float sBB[NC];
    __shared__ float sG2[NC * 2];      // gated spatial center coords
    __shared__ float sXX[8][16];       // per-wave ||x||^2
    __shared__ float sD[8][16 * 16];   // per-wave 16x16 D^2 tile scratch

    int tid  = threadIdx.x;
    int wave = tid >> 5;               // wave32
    int lane = tid & 31;
    int m    = lane & 15;
    int h    = lane >> 4;              // half-wave id (0/1)
    int base = blockIdx.x * 128 + wave * 16;

    for (int i = tid; i < NF * NC; i += 256) sBT[i] = bT[i];
    for (int i = tid; i < NC; i += 256) sBB[i] = bb[i];
    for (int i = tid; i < NC * 2; i += 256) sG2[i] = g2[i];

    // lanes 0-15: ||x||^2 for their point, keep spatial coords for the scan
    float x0 = 0.f, x1 = 0.f;
    {
        const float* xq = data + (size_t)(base + m) * NF;
        __builtin_prefetch(xq, 0, 1);          // global_prefetch_b8
        if (lane < 16) {
            float acc = 0.f;
            #pragma unroll
            for (int f = 0; f < NF; ++f) { float v = xq[f]; acc += v * v; }
            sXX[wave][m] = acc;
            x0 = xq[0]; x1 = xq[1];
        }
    }
    __syncthreads();

    // A fragments: 32-bit 16x4 layout — lane<16: K=v (v=vgpr 0/1); lane>=16: K=v+2.
    // So lane feature index f = kchunk_base + 2*h + v.
    const float* xp = data + (size_t)(base + m) * NF;
    v2f a_lo, a_hi;
    a_lo.x = xp[2 * h + 0];     a_lo.y = xp[2 * h + 1];       // K in [0,4)
    a_hi.x = xp[4 + 2 * h + 0]; a_hi.y = xp[4 + 2 * h + 1];   // K in [4,8)

    // xx values this lane will need when emitting C: row M = v + 8*h
    float xxv[8];
    #pragma unroll
    for (int v = 0; v < 8; ++v) xxv[v] = sXX[wave][v + 8 * h];

    float dval2 = 1.0e6f;   // carried scan state in squared domain (1000^2)
    int   label = -1;

    for (int j = 0; j < 8; ++j) {
        int n = j * 16 + m;   // center column this lane loads for B
        // B fragments (4x16, KxN), mirroring the A K-striping across half-waves.
        v2f b_lo, b_hi;
        b_lo.x = sBT[(2 * h + 0) * NC + n];
        b_lo.y = sBT[(2 * h + 1) * NC + n];
        b_hi.x = sBT[(4 + 2 * h + 0) * NC + n];
        b_hi.y = sBT[(4 + 2 * h + 1) * NC + n];

        v8f acc = {};
        acc = __builtin_amdgcn_wmma_f32_16x16x4_f32(false, a_lo, false, b_lo,
                                                    (short)0, acc, false, false);
        acc = __builtin_amdgcn_wmma_f32_16x16x4_f32(false, a_hi, false, b_hi,
                                                    (short)0, acc, false, false);

        // Emit D^2 = max(xx - 2*xy + bb, 0). C/D layout: M = v + 8*h, N = m.
        float bbn = sBB[n];
        #pragma unroll
        for (int v = 0; v < 8; ++v) {
            float d2 = xxv[v] - 2.f * acc[v] + bbn;
            sD[wave][(v + 8 * h) * 16 + m] = fmaxf(d2, 0.f);
        }

        // Sequential per-center scan for this tile (lanes 0-15, point = base+lane).
        // Same-wave LDS RAW: DS ops from one wave are processed in order.
        // Branch-free: gated-out centers sit at 1e30 so `active` is false.
        if (lane < 16) {
            #pragma unroll
            for (int t = 0; t < 16; ++t) {
                int i = j * 16 + t;
                float dx = x0 - sG2[2 * i];
                float dy = x1 - sG2[2 * i + 1];
                bool active = (dx * dx + dy * dy) <= r2;   // in_r && enough
                float D2 = sD[wave][lane * 16 + t];
                bool upd = active && (D2 < dval2);
                dval2 = active ? (upd ? D2 : 0.f) : dval2; // torch zeroing quirk
                label = upd ? i : label;
            }
        }
    }
    if (lane < 16) tmp_labels[base + lane] = label;
}

// -------------------- commit labels unless converged ------------------------
__global__ void spc_commit_kernel(const int* __restrict__ conv,
                                  const int* __restrict__ tmp_labels,
                                  int* __restrict__ out_labels) {
    if (*conv) return;
    int p = blockIdx.x * 256 + threadIdx.x;
    out_labels[p] = tmp_labels[p];
}

// ------------------------- segment sums -------------------------------------
__global__ void spc_reduce_kernel(const float* __restrict__ data,
                                  const int* __restrict__ tmp_labels,
                                  float* __restrict__ sums, float* __restrict__ cnts) {
    __shared__ float sS[NC * NF];
    __shared__ float sN[NC];
    int tid = threadIdx.x;
    for (int i = tid; i < NC * NF; i += 256) sS[i] = 0.f;
    for (int i = tid; i < NC; i += 256) sN[i] = 0.f;
    __syncthreads();
    int p = blockIdx.x * 256 + tid;
    int lab = tmp_labels[p];
    if (lab >= 0) {
        #pragma unroll
        for (int f = 0; f < NF; ++f)
            atomicAdd(&sS[lab * NF + f], data[(size_t)p * NF + f]);
        atomicAdd(&sN[lab], 1.f);
    }
    __syncthreads();
    for (int i = tid; i < NC * NF; i += 256) atomicAdd(&sums[i], sS[i]);
    for (int i = tid; i < NC; i += 256) atomicAdd(&cnts[i], sN[i]);
}

// ------------------------- center update ------------------------------------
__global__ void spc_update_kernel(float* __restrict__ cen,
                                  const float* __restrict__ sums,
                                  const float* __restrict__ cnts,
                                  int* __restrict__ conv) {
    __shared__ float sDiff[NC];
    __shared__ int sConv;
    int i = threadIdx.x;               // 0..127
    if (i == 0) sConv = *conv;
    __syncthreads();
    int c_in = sConv;
    float cnt = cnts[i];
    float newc[NF];
    float d2 = 0.f;
    #pragma unroll
    for (int f = 0; f < NF; ++f) {
        float oldv = cen[i * NF + f];
        float nv = (cnt > 0.f) ? (sums[i * NF + f] / fmaxf(cnt, 1.f)) : oldv;
        newc[f] = nv;
        float d = nv - oldv;
        d2 += d * d;
    }
    sDiff[i] = d2;
    __syncthreads();
    for (int s = 64; s > 0; s >>= 1) {
        if (i < s) sDiff[i] += sDiff[i + s];
        __syncthreads();
    }
    if (!c_in) {
        #pragma unroll
        for (int f = 0; f < NF; ++f) cen[i * NF + f] = newc[f];
        if (i == 0 && sqrtf(sDiff[0]) < THRESH) *conv = 1;
    }
}

// ------------------- final: centers -> output tail --------------------------
__global__ void spc_finish_kernel(const float* __restrict__ cen,
                                  float* __restrict__ out_centers) {
    int i = threadIdx.x + blockIdx.x * 256;
    if (i < NC * NF) out_centers[i] = cen[i];
}

// ---------------------------------------------------------------------------
extern "C" void kernel_launch(void* const* d_in, const int* in_sizes, int n_in,
                              void* d_out, int out_size, void* d_ws, size_t ws_size,
                              hipStream_t stream) {
    const float* data       = (const float*)d_in[0];   // [N, 8]
    const float* centers_in = (const float*)d_in[1];   // [128, 8]
    int n = in_sizes[0] / NF;                          // 524288

    float r  = 0.005f * sqrtf((float)n / (float)NC);
    float r2 = r * r;

    // Workspace layout (bytes)
    char* ws = (char*)d_ws;
    float* cen    = (float*)(ws + 0);          // 128*8 f32
    float* bT     = (float*)(ws + 4096);       // 8*128 f32 (c - eps, transposed)
    float* bb     = (float*)(ws + 8192);       // 128 f32
    int*   counts = (int*)  (ws + 8704);       // 128 i32
    float* g2     = (float*)(ws + 9216);       // 128*2 f32 gated spatial coords
    float* sums   = (float*)(ws + 10240);      // 128*8 f32
    float* cnts   = (float*)(ws + 14336);      // 128 f32
    int*   conv   = (int*)  (ws + 14848);      // 1 i32
    int*   tmpl   = (int*)  (ws + 14912);      // N i32

    // Output: labels (int32 bits) in first N slots, centers after.
    int*   out_labels  = (int*)d_out;
    float* out_centers = (float*)d_out + n;

    spc_init_kernel<<<1, 1024, 0, stream>>>(centers_in, cen, conv);

    int gridP = n / 256;   // 2048 point-parallel blocks
    int gridA = n / 128;   // 4096 assignment blocks (8 waves x 16 pts)

    for (int it = 0; it < MAX_ITERS; ++it) {
        spc_zero_counts_kernel<<<1, NC, 0, stream>>>(counts);
        spc_count_kernel<<<gridP, 256, 0, stream>>>(data, cen, counts, r2);
        spc_prep_kernel<<<1, NC, 0, stream>>>(cen, counts, bT, bb, g2, sums, cnts);
        spc_assign_kernel<<<gridA, 256, 0, stream>>>(data, g2, bT, bb, tmpl, r2);
        spc_commit_kernel<<<gridP, 256, 0, stream>>>(conv, tmpl, out_labels);
        spc_reduce_kernel<<<gridP, 256, 0, stream>>>(data, tmpl, sums, cnts);
        spc_update_kernel<<<1, NC, 0, stream>>>(cen, sums, cnts, conv);
    }
    spc_finish_kernel<<<1, 256, 0, stream>>>(cen, out_centers);
    (void)n_in; (void)out_size; (void)ws_size;
}